// InsideLoss_42769284334027
// MI455X (gfx1250) — compile-verified
//
#include <hip/hip_runtime.h>

typedef __attribute__((ext_vector_type(2))) float v2f;
typedef __attribute__((ext_vector_type(8))) float v8f;

#define BN   16            // number of pred boxes
#define FN   16            // number of fragment boxes
#define FPTS 400           // FP: grid samples per pred box
#define BPTS 400           // BP: boundary samples per fragment box
#define MTOT (BN * FPTS)   // 6400 point rows
#define NTOT (FN * BPTS)   // 6400 boundary cols
#define COLT (BPTS / 16)   // 25 column tiles per fragment
#define MT   (MTOT / 16)   // 400 row tiles
#define NPAIR (MT * FN)    // 6400 (row-tile, fragment) wave tasks

// Workspace layout (floats):
//   P      [0,          2*MTOT)   point xy interleaved
//   pnorm  [2*MTOT,     3*MTOT)
//   Q      [3*MTOT,     5*MTOT)   boundary xy interleaved
//   qnorm  [5*MTOT,     6*MTOT)
//   partial[6*MTOT,     7*MTOT)   per-wave partial sums

__global__ void InsideLoss_prep_kernel(const float* __restrict__ pred,
                                       const float* __restrict__ fboxes,
                                       const float* __restrict__ boundary,
                                       const float* __restrict__ fragment,
                                       float* __restrict__ ws) {
  int tid = blockIdx.x * blockDim.x + threadIdx.x;
  float* P     = ws;
  float* pnorm = ws + 2 * MTOT;
  float* Q     = ws + 3 * MTOT;
  float* qnorm = ws + 5 * MTOT;
  if (tid < MTOT) {
    int b = tid / FPTS, p = tid % FPTS;
    float x1 = pred[4 * b + 0], y1 = pred[4 * b + 1];
    float wx = pred[4 * b + 2] - x1, wy = pred[4 * b + 3] - y1;
    float px = fragment[2 * p] * wx + x1;
    float py = fragment[2 * p + 1] * wy + y1;
    P[2 * tid]     = px;
    P[2 * tid + 1] = py;
    pnorm[tid] = px * px + py * py;
  } else if (tid < 2 * MTOT) {
    int n = tid - MTOT;
    int f = n / BPTS, k = n % BPTS;
    float x1 = fboxes[4 * f + 0], y1 = fboxes[4 * f + 1];
    float wx = fboxes[4 * f + 2] - x1, wy = fboxes[4 * f + 3] - y1;
    float qx = boundary[2 * k] * wx + x1;
    float qy = boundary[2 * k + 1] * wy + y1;
    Q[2 * n]     = qx;
    Q[2 * n + 1] = qy;
    qnorm[n] = qx * qx + qy * qy;
  }
}

// One wave32 per (16-row point tile, fragment) pair.
// dist^2 = pnorm + qnorm - 2*p.q ; p.q via V_WMMA_F32_16X16X4_F32 (K padded 2->4).
__global__ void __launch_bounds__(256)
InsideLoss_wmma_kernel(const float* __restrict__ fboxes,
                       const float* __restrict__ ws,
                       float* __restrict__ partial) {
  const float* P     = ws;
  const float* pnorm = ws + 2 * MTOT;
  const float* Q     = ws + 3 * MTOT;
  const float* qnorm = ws + 5 * MTOT;

  int wave = (blockIdx.x * blockDim.x + threadIdx.x) >> 5;
  int lane = threadIdx.x & 31;
  int tile_m = wave >> 4;   // / FN
  int f      = wave & 15;   // % FN

  bool lo  = lane < 16;
  int  l15 = lane & 15;

  // A tile (16x4 f32): lanes 0-15 hold K=0,1 (x,y) of row M=lane;
  // lanes 16-31 hold the K=2,3 zero padding.
  v2f a;
  int m = tile_m * 16 + l15;
  if (lo) {
    a.x = P[2 * m];
    a.y = P[2 * m + 1];
  } else {
    a.x = 0.0f;
    a.y = 0.0f;
  }

  v8f acc;
#pragma unroll
  for (int v = 0; v < 8; ++v) acc[v] = 3.4e38f;

  int nbase = f * BPTS;
  for (int kt = 0; kt < COLT; ++kt) {
    int n = nbase + kt * 16 + l15;
    // B tile (4x16 f32): lanes 0-15 hold K=0,1 rows (x,y of col N=lane);
    // lanes 16-31 hold the K=2,3 zero rows.
    v2f bv;
    if (lo) {
      bv.x = Q[2 * n];
      bv.y = Q[2 * n + 1];
    } else {
      bv.x = 0.0f;
      bv.y = 0.0f;
    }
    float qn = qnorm[n];

    v8f c = {0.f, 0.f, 0.f, 0.f, 0.f, 0.f, 0.f, 0.f};
    v8f d = __builtin_amdgcn_wmma_f32_16x16x4_f32(
        /*neg_a=*/false, a, /*neg_b=*/false, bv,
        /*c_mod=*/(short)0, c, /*reuse_a=*/false, /*reuse_b=*/false);

#pragma unroll
    for (int v = 0; v < 8; ++v) {
      float t = qn - 2.0f * d[v];      // pnorm added after the min
      acc[v] = fminf(acc[v], t);
    }
  }

  // Per-row min: butterfly over the 16 lanes holding that row's columns.
#pragma unroll
  for (int v = 0; v < 8; ++v) {
    float t = acc[v];
    t = fminf(t, __shfl_xor(t, 1, 32));
    t = fminf(t, __shfl_xor(t, 2, 32));
    t = fminf(t, __shfl_xor(t, 4, 32));
    t = fminf(t, __shfl_xor(t, 8, 32));
    acc[v] = t;
  }

  // Epilogue: outside-box mask and partial sum.
  float bx1 = fboxes[4 * f + 0], by1 = fboxes[4 * f + 1];
  float bx2 = fboxes[4 * f + 2], by2 = fboxes[4 * f + 3];
  int rbase = tile_m * 16 + (lo ? 0 : 8);
  float sum = 0.0f;
#pragma unroll
  for (int v = 0; v < 8; ++v) {
    int r = rbase + v;
    float px = P[2 * r], py = P[2 * r + 1];
    float dist = acc[v] + pnorm[r];
    bool inside = (px >= bx1) && (py >= by1) && (bx2 >= px) && (by2 >= py);
    sum += inside ? 0.0f : dist;
  }
  // lanes 0-15 all hold the rows 0-7 sum, lanes 16-31 the rows 8-15 sum.
  sum = sum + __shfl_xor(sum, 16, 32);
  if (lane == 0) partial[wave] = sum;
}

__global__ void InsideLoss_reduce_kernel(const float* __restrict__ partial,
                                         float* __restrict__ out) {
  __shared__ float s[256];
  int t = threadIdx.x;
  float acc = 0.0f;
  for (int i = t; i < NPAIR; i += 256) acc += partial[i];
  s[t] = acc;
  __syncthreads();
  for (int off = 128; off > 0; off >>= 1) {
    if (t < off) s[t] += s[t + off];
    __syncthreads();
  }
  if (t == 0) out[0] = s[0] / (float)MTOT;   // / (B * FP)
}

extern "C" void kernel_launch(void* const* d_in, const int* in_sizes, int n_in,
                              void* d_out, int out_size, void* d_ws, size_t ws_size,
                              hipStream_t stream) {
  const float* pred     = (const float*)d_in[0];
  const float* fboxes   = (const float*)d_in[1];
  const float* boundary = (const float*)d_in[2];
  const float* fragment = (const float*)d_in[3];
  float* ws      = (float*)d_ws;
  float* partial = ws + 6 * MTOT;

  InsideLoss_prep_kernel<<<(2 * MTOT + 255) / 256, 256, 0, stream>>>(
      pred, fboxes, boundary, fragment, ws);
  // 6400 waves, 8 waves (256 threads) per block -> 800 blocks
  InsideLoss_wmma_kernel<<<NPAIR / 8, 256, 0, stream>>>(fboxes, ws, partial);
  InsideLoss_reduce_kernel<<<1, 256, 0, stream>>>(partial, (float*)d_out);
}